// CTCLossFromScratch_88983132439264
// MI455X (gfx1250) — compile-verified
//
#include <hip/hip_runtime.h>
#include <math.h>

// CTC forward loss, MI455X (gfx1250). Single wave32 per batch element,
// barrier-free recursion via lane shuffles, async-DMA emission staging.
// Shapes fixed by the reference: T=256, B=64, C=8000, L_MAX=32, S=65.
#define T_MAXV 256
#define BN     64
#define CN     8000
#define LMAXN  32
#define SN     65          // 2*L_MAX + 1
#define CHN    16          // timesteps per async chunk (32 loads/chunk/wave)
#define NBUF   3           // LDS ring depth (prefetch distance = 2 chunks)
#define NEGV   (-1e9f)

// logaddexp matching jnp.logaddexp: max + log1p(exp(-|d|)).
// (NEG,NEG): d=0 -> NEG + log(2) rounds back to NEG in f32 (ulp ~64).
__device__ __forceinline__ float lae(float a, float b) {
    float mx = fmaxf(a, b);
    float mn = fminf(a, b);
    return mx + log1pf(__expf(mn - mx));
}

// Low 32 bits of a generic LDS pointer == wave-relative LDS byte offset.
__device__ __forceinline__ unsigned lds_off(const void* p) {
    return (unsigned)(uintptr_t)p;
}

// Per-lane gather: copies 4 bytes from the global address sbase + goff to the LDS offset in loff. Tracked by ASYNCcnt.
__device__ __forceinline__ void async_gather_b32(unsigned loff, unsigned goff,
                                                 const void* sbase) {
    asm volatile("global_load_async_to_lds_b32 %0, %1, %2"
                 :: "v"(loff), "v"(goff), "s"(sbase)
                 : "memory");
}

// Extended-target label for state s (blank-interleaved, blank-padded).
__device__ __forceinline__ int ext_label(const int* __restrict__ tg, int tl, int s) {
    if (!(s & 1)) return 0;
    int j = (s - 1) >> 1;
    return (j < tl) ? tg[j] : 0;
}

__global__ __launch_bounds__(32)
void ctc_alpha_kernel(const float* __restrict__ logp,    // [T,B,C] log-softmaxed
                      const int*   __restrict__ targets, // [B,L_MAX]
                      const int*   __restrict__ in_len,  // [B]
                      const int*   __restrict__ tg_len,  // [B]
                      float*       __restrict__ ws)      // [B] normalized losses
{
    __shared__ float em[NBUF][CHN][64];   // staged emissions for states 0..63

    const int b    = blockIdx.x;
    const int lane = threadIdx.x;         // 0..31, one full wave32
    const int il   = in_len[b];           // uniform
    const int tl   = tg_len[b];           // uniform
    const int* tg  = targets + b * LMAXN;

    // ---- per-lane labels & skip flags for the two owned states ----
    const int s_lo = lane;                // states 0..31
    const int s_hi = lane + 32;           // states 32..63  (state 64 is uniform)
    const int lab_lo  = ext_label(tg, tl, s_lo);
    const int lab_hi  = ext_label(tg, tl, s_hi);
    const int lab2_lo = (s_lo >= 2) ? ext_label(tg, tl, s_lo - 2) : -1;
    const int lab2_hi = ext_label(tg, tl, s_hi - 2);
    const bool skip_lo = (lab_lo != 0) && (lab_lo != lab2_lo);
    const bool skip_hi = (lab_hi != 0) && (lab_hi != lab2_hi);

    const char*  gbase     = (const char*)logp + (size_t)b * (size_t)CN * 4u;
    const size_t rowstride = (size_t)BN * (size_t)CN * 4u;   // bytes per t-step
    const unsigned goff_lo = (unsigned)lab_lo * 4u;
    const unsigned goff_hi = (unsigned)lab_hi * 4u;

    const int nch = (il + CHN - 1) / CHN;   // <= 16; always issue FULL chunks
                                            // (rows up to nch*16 <= 256, in-bounds)

    // ---- issue one full chunk (32 async loads per wave) ----
    auto issue_chunk = [&](int c) {
        const int buf = c % NBUF;
        const char* gb = gbase + (size_t)(c * CHN) * rowstride;
        unsigned loff0 = lds_off(&em[buf][0][s_lo]);
        unsigned loff1 = lds_off(&em[buf][0][s_hi]);
        #pragma unroll
        for (int tt = 0; tt < CHN; ++tt) {
            async_gather_b32(loff0, goff_lo, gb);
            async_gather_b32(loff1, goff_hi, gb);
            loff0 += 64u * 4u;
            loff1 += 64u * 4u;
            gb    += rowstride;
        }
    };

    issue_chunk(0);
    if (nch > 1) issue_chunk(1);

    float a_lo = NEGV, a_hi = NEGV, a_top = NEGV;   // alpha registers

    for (int k = 0; k < nch; ++k) {
        // Prefetch 2 chunks ahead, then wait for chunk k.
        // In-order completion: after issuing chunk k+2 there are 64 younger
        // loads; outstanding <= 63 forces chunk k complete.
        if (k + 2 < nch) {
            issue_chunk(k + 2);
            asm volatile("s_wait_asynccnt 0x3f" ::: "memory");
        } else if (k + 1 < nch) {
            asm volatile("s_wait_asynccnt 0x20" ::: "memory");
        } else {
            asm volatile("s_wait_asynccnt 0x0" ::: "memory");
        }

        const int buf  = k % NBUF;
        const int t0   = k * CHN;
        int tend = il - t0; if (tend > CHN) tend = CHN;   // compute only t < il
        int tt0  = 0;

        if (k == 0) {
            // t=0 init: a0[0]=lp0[blank]=em[0][0][0]; a0[1]=lp0[ext1]=em[0][0][1]
            float e0 = em[0][0][s_lo];
            a_lo = ((lane == 0) || (lane == 1 && tl > 0)) ? e0 : NEGV;
            a_hi = NEGV;
            a_top = NEGV;
            tt0 = 1;
        }

        for (int tt = tt0; tt < tend; ++tt) {
            const float em_lo  = em[buf][tt][s_lo];
            const float em_hi  = em[buf][tt][s_hi];
            const float em_top = __shfl(em_lo, 0, 32);   // state 64 is blank == state 0

            // neighbor alphas via wave shuffles (no barriers, no LDS)
            float lo_m1 = __shfl_up(a_lo, 1, 32);
            float lo_m2 = __shfl_up(a_lo, 2, 32);
            float hi_m1 = __shfl_up(a_hi, 1, 32);
            float hi_m2 = __shfl_up(a_hi, 2, 32);
            const float lo30 = __shfl(a_lo, 30, 32);
            const float lo31 = __shfl(a_lo, 31, 32);
            const float hi31 = __shfl(a_hi, 31, 32);
            if (lane == 0) { lo_m1 = NEGV; lo_m2 = NEGV; hi_m1 = lo31; hi_m2 = lo30; }
            if (lane == 1) { lo_m2 = NEGV; hi_m2 = lo31; }

            float p_lo = lae(a_lo, lo_m1);
            if (skip_lo) p_lo = lae(p_lo, lo_m2);
            float p_hi = lae(a_hi, hi_m1);
            if (skip_hi) p_hi = lae(p_hi, hi_m2);
            float p_top = lae(a_top, hi31);              // state 64: blank, no skip

            a_lo  = p_lo  + em_lo;
            a_hi  = p_hi  + em_hi;
            a_top = p_top + em_top;
        }
    }

    // ---- finalize: loss[b] / max(tl,1) ----
    const int ib = 2 * tl;
    int ic = 2 * tl - 1; if (ic < 0) ic = 0;
    auto get_alpha = [&](int s) -> float {
        if (s == 64) return a_top;
        if (s >= 32) return __shfl(a_hi, s - 32, 32);
        return __shfl(a_lo, s, 32);
    };
    const float vb = get_alpha(ib);
    const float vc = get_alpha(ic);
    if (lane == 0) {
        const float v  = (tl > 0) ? lae(vb, vc) : vb;
        const int   sl = (tl > 0) ? tl : 1;
        ws[b] = (-v) / (float)sl;
    }
}

// Deterministic sequential mean over B values.
__global__ void ctc_reduce_kernel(const float* __restrict__ ws,
                                  float* __restrict__ out, int n) {
    if (threadIdx.x == 0 && blockIdx.x == 0) {
        float s = 0.0f;
        for (int i = 0; i < n; ++i) s += ws[i];
        out[0] = s / (float)n;
    }
}

extern "C" void kernel_launch(void* const* d_in, const int* in_sizes, int n_in,
                              void* d_out, int out_size, void* d_ws, size_t ws_size,
                              hipStream_t stream) {
    const float* logp    = (const float*)d_in[0];
    const int*   targets = (const int*)  d_in[1];
    const int*   in_len  = (const int*)  d_in[2];
    const int*   tg_len  = (const int*)  d_in[3];

    const int Bn = in_sizes[2];                    // == 64
    float* ws  = (float*)d_ws;
    float* out = (float*)d_out;

    ctc_alpha_kernel<<<Bn, 32, 0, stream>>>(logp, targets, in_len, tg_len, ws);
    ctc_reduce_kernel<<<1, 32, 0, stream>>>(ws, out, Bn);
}